// AdaptiveNormalization_70257075028068
// MI455X (gfx1250) — compile-verified
//
#include <hip/hip_runtime.h>
#include <math.h>

typedef __attribute__((ext_vector_type(2))) float v2f;
typedef __attribute__((ext_vector_type(4))) float f4;
typedef __attribute__((ext_vector_type(8))) float v8f;

#define MOM   0.01f
#define OM    0.99f
#define EPSV  1e-6f
#define CCH   256
#define TT    16384
#define CHUNK 64            // TT / 256 threads

// ---------------------------------------------------------------------------
// Kernel A: per-(b,t) channel reductions via V_WMMA_F32_16X16X4_F32.
// One wave = one (batch, 16-wide t tile). Accumulate over C in K=4 steps.
//   A(16x4): row0 = w_shift[k], row1 = exp(w_scale_log)[k], rows 2..15 = 0
//   B(4x16): x[c0+K][t0+N]      (second chain: x^2)
//   D(16x16): row0 = s, row1 = Q1 (x-chain); row1 = Q2 (x^2-chain)
// VGPR layouts (ISA 7.12.2, 32-bit):
//   A: lane%16 = M, K = 2*(lane/16) + vgpr
//   B: lane%16 = N, K = 2*(lane/16) + vgpr
//   D: lane%16 = N, M = vgpr + 8*(lane/16)
// A-fragment values come from ONE combined LDS table indexed by a per-lane
// row base computed before the loop -> branchless ds_load_b64 per iteration.
// ---------------------------------------------------------------------------
__global__ __launch_bounds__(256) void ka_reduce(
    const float* __restrict__ x, const float* __restrict__ w_shift,
    const float* __restrict__ w_scale_log,
    float* __restrict__ sArr, float* __restrict__ q1Arr, float* __restrict__ q2Arr)
{
  __shared__ float wAB[3 * CCH];  // [0:C)=w_shift, [C:2C)=ew, [2C:3C)=0
  int tid = threadIdx.x;
  wAB[tid]           = w_shift[tid];
  wAB[CCH + tid]     = expf(w_scale_log[tid]);
  wAB[2 * CCH + tid] = 0.f;
  __syncthreads();

  int lane = tid & 31;
  int wave = tid >> 5;
  int n = lane & 15;          // N (B/D column) and M (A row) selector
  int h = lane >> 4;          // lane half -> K pair
  int tile = blockIdx.x * 8 + wave;
  int bb = tile >> 10;                    // tile / (TT/16)
  int t0 = (tile & 1023) << 4;
  const float* xb = x + (size_t)bb * CCH * TT + (size_t)t0 + n;

  int rowBase = ((n < 2) ? n : 2) * CCH;  // per-lane A-row table base

  v8f accW = {0.f,0.f,0.f,0.f,0.f,0.f,0.f,0.f};
  v8f accQ = {0.f,0.f,0.f,0.f,0.f,0.f,0.f,0.f};

  #pragma unroll 8
  for (int c0 = 0; c0 < CCH; c0 += 4) {
    int k = c0 + 2 * h;
    float b0 = xb[(size_t)k * TT];
    float b1 = xb[(size_t)(k + 1) * TT];
    v2f Av = { wAB[rowBase + k], wAB[rowBase + k + 1] };  // one ds_load_b64
    v2f Bv = { b0, b1 };
    v2f Bq = { b0 * b0, b1 * b1 };
    accW = __builtin_amdgcn_wmma_f32_16x16x4_f32(false, Av, false, Bv,
                                                 (short)0, accW, false, false);
    accQ = __builtin_amdgcn_wmma_f32_16x16x4_f32(false, Av, false, Bq,
                                                 (short)0, accQ, false, false);
  }
  if (lane < 16) {  // D rows 0,1 live in vgprs 0,1 of lanes 0..15
    size_t o = (size_t)bb * TT + t0 + n;
    sArr[o]  = accW[0];   // row0: w_shift . x
    q1Arr[o] = accW[1];   // row1: ew . x
    q2Arr[o] = accQ[1];   // row1: ew . x^2
  }
}

// ---------------------------------------------------------------------------
// Kernel B: one block per batch. Chunked parallel scan of the two EMAs.
// y_end(chunk) = F + D * y_in  is linear -> associative (F,D) combine.
// ---------------------------------------------------------------------------
__global__ __launch_bounds__(256) void kb_scan(
    const float* __restrict__ sArr, const float* __restrict__ q1Arr,
    const float* __restrict__ q2Arr, const float* __restrict__ w_scale_log,
    float* __restrict__ shiftArr, float* __restrict__ invArr)
{
  __shared__ float scF[256];
  __shared__ float scD[256];
  __shared__ float red[256];
  int tid = threadIdx.x;
  size_t base = (size_t)blockIdx.x * TT + (size_t)tid * CHUNK;

  // W = sum_c exp(w_scale_log[c])
  red[tid] = expf(w_scale_log[tid]);
  __syncthreads();
  for (int o = 128; o > 0; o >>= 1) {
    if (tid < o) red[tid] += red[tid + o];
    __syncthreads();
  }
  float Wsum = red[0];
  __syncthreads();

  float dchunk = 1.f;
  #pragma unroll
  for (int i = 0; i < CHUNK; ++i) dchunk *= OM;   // (1-m)^CHUNK, folded

  // ---- phase 1: chunk-local EMA final of s (zero init) ----
  float y = 0.f;
  #pragma unroll 4
  for (int j = 0; j < CHUNK; ++j) y = MOM * sArr[base + j] + OM * y;

  // ---- scan 1 over (F,D) carries ----
  scF[tid] = y; scD[tid] = dchunk;
  __syncthreads();
  for (int ofs = 1; ofs < 256; ofs <<= 1) {
    float f1 = 0.f, d1 = 1.f;
    if (tid >= ofs) { f1 = scF[tid - ofs]; d1 = scD[tid - ofs]; }
    float f2 = scF[tid], d2 = scD[tid];
    __syncthreads();
    scF[tid] = f2 + d2 * f1;
    scD[tid] = d2 * d1;
    __syncthreads();
  }
  float carry = (tid > 0) ? scF[tid - 1] : 0.f;
  __syncthreads();

  // ---- phase 2: replay shift, form q, chunk-local EMA final of q ----
  float qreg[CHUNK];
  y = carry;
  float g = 0.f;
  #pragma unroll
  for (int j = 0; j < CHUNK; ++j) {
    float sv = sArr[base + j];
    y = MOM * sv + OM * y;
    shiftArr[base + j] = y;
    float q = q2Arr[base + j] - 2.f * y * q1Arr[base + j] + y * y * Wsum;
    qreg[j] = q;
    g = MOM * q + OM * g;
  }

  // ---- scan 2 ----
  scF[tid] = g; scD[tid] = dchunk;
  __syncthreads();
  for (int ofs = 1; ofs < 256; ofs <<= 1) {
    float f1 = 0.f, d1 = 1.f;
    if (tid >= ofs) { f1 = scF[tid - ofs]; d1 = scD[tid - ofs]; }
    float f2 = scF[tid], d2 = scD[tid];
    __syncthreads();
    scF[tid] = f2 + d2 * f1;
    scD[tid] = d2 * d1;
    __syncthreads();
  }
  float carry2 = (tid > 0) ? scF[tid - 1] : 0.f;

  // ---- phase 3: replay scale, emit 1/sqrt(scale+eps) ----
  float y2 = carry2;
  #pragma unroll
  for (int j = 0; j < CHUNK; ++j) {
    y2 = MOM * qreg[j] + OM * y2;
    invArr[base + j] = 1.f / sqrtf(y2 + EPSV);
  }
}

// ---------------------------------------------------------------------------
// Kernel C: out = (x - shift) * inv * w_proj + b_proj   (float4, NT stores)
// Each block covers 1024 contiguous t within one (b,c) row -> c is uniform.
// x (134 MB) was read by kernel A and fits in the 192 MB L2, so this pass
// re-reads it at L2 bandwidth; NT stores keep `out` from thrashing L2.
// ---------------------------------------------------------------------------
__global__ __launch_bounds__(256) void kc_apply(
    const float* __restrict__ x, const float* __restrict__ w_proj,
    const float* __restrict__ b_proj, const float* __restrict__ shiftArr,
    const float* __restrict__ invArr, float* __restrict__ out)
{
  size_t flat = ((size_t)blockIdx.x * 256 + threadIdx.x) * 4;
  int row = (int)(flat / TT);          // b*C + c
  int t   = (int)(flat & (TT - 1));
  int c   = row & (CCH - 1);
  int bb  = row >> 8;
  f4 xv = *(const f4*)(x + flat);
  size_t so = (size_t)bb * TT + t;
  f4 sh = *(const f4*)(shiftArr + so);
  f4 iv = *(const f4*)(invArr + so);
  float wp = w_proj[c], bp = b_proj[c];
  f4 o;
  o.x = (xv.x - sh.x) * iv.x * wp + bp;
  o.y = (xv.y - sh.y) * iv.y * wp + bp;
  o.z = (xv.z - sh.z) * iv.z * wp + bp;
  o.w = (xv.w - sh.w) * iv.w * wp + bp;
  __builtin_nontemporal_store(o, (f4*)(out + flat));
}

// ---------------------------------------------------------------------------
extern "C" void kernel_launch(void* const* d_in, const int* in_sizes, int n_in,
                              void* d_out, int out_size, void* d_ws, size_t ws_size,
                              hipStream_t stream) {
  const float* x            = (const float*)d_in[0];
  const float* w_shift      = (const float*)d_in[1];
  const float* w_scale_log  = (const float*)d_in[2];
  const float* w_proj       = (const float*)d_in[3];
  const float* b_proj       = (const float*)d_in[4];
  float* out = (float*)d_out;

  int Bn = in_sizes[0] / (CCH * TT);            // 8
  size_t BT = (size_t)Bn * TT;
  float* ws       = (float*)d_ws;               // 5*BT floats (~2.6 MB)
  float* sArr     = ws;
  float* q1Arr    = ws + BT;
  float* q2Arr    = ws + 2 * BT;
  float* shiftArr = ws + 3 * BT;
  float* invArr   = ws + 4 * BT;

  // A: Bn*(T/16) tiles, 8 waves (tiles) per 256-thread block
  ka_reduce<<<Bn * (TT / 16) / 8, 256, 0, stream>>>(x, w_shift, w_scale_log,
                                                    sArr, q1Arr, q2Arr);
  // B: one block per batch
  kb_scan<<<Bn, 256, 0, stream>>>(sArr, q1Arr, q2Arr, w_scale_log,
                                  shiftArr, invArr);
  // C: elementwise, float4 per thread
  size_t elems = (size_t)in_sizes[0];
  kc_apply<<<(int)(elems / 1024), 256, 0, stream>>>(x, w_proj, b_proj,
                                                    shiftArr, invArr, out);
}